// KSparseAutoencoder_41291815584091
// MI455X (gfx1250) — compile-verified
//
#include <hip/hip_runtime.h>
#include <hip/hip_bf16.h>

// ---------------------------------------------------------------------------
// k-Sparse Autoencoder for MI455X (gfx1250)
//   B=4096, D=768, H=16384, K=32
//   out = [ x_hat (B*D) | z_masked (B*H) ]  (fp32)
// ---------------------------------------------------------------------------

typedef __attribute__((ext_vector_type(2))) float v2f;
typedef __attribute__((ext_vector_type(4))) float v4f;
typedef __attribute__((ext_vector_type(8))) float v8f;

#define B_DIM 4096
#define D_DIM 768
#define H_DIM 16384
#define K_TOP 32

// ======================= 1) Encode GEMM (fp32 WMMA) ========================
// z[b,h] = ReLU( dot(x[b,:], W_enc[h,:]) + b_enc[h] )
// Block tile 64(M) x 128(N), 4 waves; each wave: 64M x 32N via 4x2 v8f accums
// (A fragments reused across two B fragments -> 0.75 ds_load_b64 per WMMA).
// Global->LDS staging is software-pipelined: next K-chunk is prefetched into
// registers while the WMMA stream consumes the current LDS tiles.
// LDS row stride padded to 36 floats: conflict-free b64 fragment reads and
// 16B-aligned float4 staging stores.
#define TM 64
#define TN 128
#define TKC 32
#define LSTR 36

__global__ __launch_bounds__(128) void encode_wmma_kernel(
    const float* __restrict__ x, const float* __restrict__ We,
    const float* __restrict__ be, float* __restrict__ z) {
  __shared__ float Xs[TM * LSTR];   // 64 x 36
  __shared__ float Ws[TN * LSTR];   // 128 x 36

  const int tid  = threadIdx.x;
  const int lane = tid & 31;
  const int wave = tid >> 5;                 // 0..3 -> N slice
  const int rowA0 = blockIdx.x * TM;         // B offset
  const int rowB0 = blockIdx.y * TN;         // H offset

  // staging maps
  const int xrow = tid >> 1;                 // 0..63
  const int xcol = (tid & 1) * 16;           // 0 / 16
  // W: thread t stages full row t (32 floats)

  const int lr    = lane & 15;               // N / M sub-index
  const int halfk = (lane >> 4) * 2;         // lanes 0-15 -> K0/K1, 16-31 -> K2/K3

  v8f acc[4][2] = {};
  v4f px[4], pw[8];

  // prefetch first K-chunk
#pragma unroll
  for (int c = 0; c < 4; ++c)
    px[c] = *(const v4f*)(x + (size_t)(rowA0 + xrow) * D_DIM + xcol + c * 4);
#pragma unroll
  for (int c = 0; c < 8; ++c)
    pw[c] = *(const v4f*)(We + (size_t)(rowB0 + tid) * D_DIM + c * 4);

  for (int k0 = 0; k0 < D_DIM; k0 += TKC) {
    __syncthreads();
#pragma unroll
    for (int c = 0; c < 4; ++c) *(v4f*)&Xs[xrow * LSTR + xcol + c * 4] = px[c];
#pragma unroll
    for (int c = 0; c < 8; ++c) *(v4f*)&Ws[tid * LSTR + c * 4] = pw[c];
    __syncthreads();

    const int kn = k0 + TKC;
    if (kn < D_DIM) {   // prefetch next chunk; overlaps with WMMA stream below
#pragma unroll
      for (int c = 0; c < 4; ++c)
        px[c] = *(const v4f*)(x + (size_t)(rowA0 + xrow) * D_DIM + kn + xcol + c * 4);
#pragma unroll
      for (int c = 0; c < 8; ++c)
        pw[c] = *(const v4f*)(We + (size_t)(rowB0 + tid) * D_DIM + kn + c * 4);
    }

#pragma unroll
    for (int k2 = 0; k2 < TKC / 4; ++k2) {
      const int col = k2 * 4 + halfk;
      v2f b0 = *(const v2f*)&Ws[(wave * 32 + lr) * LSTR + col];
      v2f b1 = *(const v2f*)&Ws[(wave * 32 + 16 + lr) * LSTR + col];
#pragma unroll
      for (int m = 0; m < 4; ++m) {
        v2f a = *(const v2f*)&Xs[(m * 16 + lr) * LSTR + col];
        acc[m][0] = __builtin_amdgcn_wmma_f32_16x16x4_f32(
            false, a, false, b0, (short)0, acc[m][0], false, false);
        acc[m][1] = __builtin_amdgcn_wmma_f32_16x16x4_f32(
            false, a, false, b1, (short)0, acc[m][1], false, false);
      }
    }
  }

  // epilogue: bias + ReLU, store unmasked z
  const int   n0    = rowB0 + wave * 32 + lr;
  const float bias0 = be[n0];
  const float bias1 = be[n0 + 16];
  const int   rbase = (lane < 16) ? 0 : 8;   // C layout: VGPR j -> M=j (+8 upper half)
#pragma unroll
  for (int m = 0; m < 4; ++m) {
#pragma unroll
    for (int j = 0; j < 8; ++j) {
      const size_t row = (size_t)(rowA0 + m * 16 + rbase + j);
      float v0 = acc[m][0][j] + bias0;
      float v1 = acc[m][1][j] + bias1;
      z[row * H_DIM + n0]      = v0 > 0.f ? v0 : 0.f;
      z[row * H_DIM + n0 + 16] = v1 > 0.f ? v1 : 0.f;
    }
  }
}

// ======================= 2) Per-row top-K selection ========================
// Radix selection on positive-float bit patterns (monotonic for z >= 0).
// One 256-thread block per row. Deterministic output: the compact (idx,val)
// list is built in ascending-index order via a block prefix scan, and ties at
// the threshold value are broken by smaller index (matching jax.lax.top_k).
#define CAND_CAP 1024

__global__ __launch_bounds__(256) void topk_mask_kernel(
    float* __restrict__ z, int* __restrict__ tidx, float* __restrict__ tval) {
  const int b = blockIdx.x;
  float* row = z + (size_t)b * H_DIM;

  __shared__ int   hist[4096];
  __shared__ float candVal[CAND_CAP];
  __shared__ int   candIdx[CAND_CAP];
  __shared__ int   nCand, nSelCand, hbinS, rS;
  __shared__ int   selIdx[K_TOP];
  __shared__ int   scan[256];

  const int tid = threadIdx.x;
  for (int i = tid; i < 4096; i += 256) hist[i] = 0;
  if (tid == 0) { nCand = 0; nSelCand = 0; }
  __syncthreads();

  // P1: histogram of high bits (sign0|exp8|mant3)
  for (int i = tid; i < H_DIM; i += 256) {
    float v = row[i];
    if (v > 0.f) atomicAdd(&hist[__float_as_uint(v) >> 20], 1);
  }
  __syncthreads();

  if (tid == 0) {
    int acc = 0, hb = 0, r = K_TOP;
    for (int bin = 4095; bin >= 0; --bin) {
      int c = hist[bin];
      if (acc + c >= K_TOP) { hb = bin; r = K_TOP - acc; break; }
      acc += c;
    }
    hbinS = hb; rS = r;
  }
  __syncthreads();
  const int hbin = hbinS, r = rS;

  // P2: gather threshold-bin candidates
  for (int i = tid; i < H_DIM; i += 256) {
    float v = row[i];
    if (v > 0.f && (int)(__float_as_uint(v) >> 20) == hbin) {
      int p = atomicAdd(&nCand, 1);
      if (p < CAND_CAP) { candVal[p] = v; candIdx[p] = i; }
    }
  }
  __syncthreads();

  // exact rank among candidates (value desc, index asc tie-break)
  int n = nCand; if (n > CAND_CAP) n = CAND_CAP;
  for (int t = tid; t < n; t += 256) {
    float v = candVal[t]; int idx = candIdx[t];
    int rank = 0;
    for (int j = 0; j < n; ++j) {
      float vj = candVal[j];
      rank += (vj > v) || (vj == v && candIdx[j] < idx);
    }
    if (rank < r) {
      int p = atomicAdd(&nSelCand, 1);
      if (p < K_TOP) selIdx[p] = idx;
    } else {
      row[idx] = 0.f;   // below threshold: zero out
    }
  }
  __syncthreads();
  int ns = nSelCand; if (ns > K_TOP) ns = K_TOP;

  // P3: zero everything below threshold, build deterministic compact list
  int   li[K_TOP]; float lv[K_TOP]; int cnt = 0;
  const int chunk = H_DIM / 256;             // 64 contiguous elems per thread
  const int base  = tid * chunk;
  for (int e = 0; e < chunk; ++e) {
    const int i = base + e;
    float v = row[i];
    if (v > 0.f) {
      const int bin = (int)(__float_as_uint(v) >> 20);
      bool sel = bin > hbin;
      if (bin == hbin) {
        for (int s = 0; s < ns; ++s)
          if (selIdx[s] == i) { sel = true; break; }
      }
      if (sel) { if (cnt < K_TOP) { li[cnt] = i; lv[cnt] = v; cnt++; } }
      else row[i] = 0.f;
    }
  }
  if (cnt > K_TOP) cnt = K_TOP;
  scan[tid] = cnt;
  __syncthreads();
  for (int off = 1; off < 256; off <<= 1) {   // Hillis-Steele inclusive scan
    int a = (tid >= off) ? scan[tid - off] : 0;
    int v = scan[tid];
    __syncthreads();
    scan[tid] = v + a;
    __syncthreads();
  }
  const int excl  = scan[tid] - cnt;
  const int total = scan[255];
  for (int j = 0; j < cnt; ++j) {
    int pos = excl + j;
    if (pos < K_TOP) {
      tidx[(size_t)b * K_TOP + pos] = li[j];
      tval[(size_t)b * K_TOP + pos] = lv[j];
    }
  }
  for (int p = total + tid; p < K_TOP; p += 256) {  // pad (harmless zeros)
    tidx[(size_t)b * K_TOP + p] = 0;
    tval[(size_t)b * K_TOP + p] = 0.f;
  }
}

// ================= 3) W_dec transpose [D,H] -> [H,D] in ws =================
__global__ __launch_bounds__(256) void transpose_kernel(
    const float* __restrict__ W, float* __restrict__ WT) {
  __shared__ float t[32][33];
  const int hx = blockIdx.x * 32;
  const int dy = blockIdx.y * 32;
  const int tx = threadIdx.x, ty = threadIdx.y;   // block (32,8)
#pragma unroll
  for (int i = 0; i < 32; i += 8)
    t[ty + i][tx] = W[(size_t)(dy + ty + i) * H_DIM + hx + tx];
  __syncthreads();
#pragma unroll
  for (int i = 0; i < 32; i += 8)
    WT[(size_t)(hx + ty + i) * D_DIM + dy + tx] = t[tx][ty + i];
}

// ===================== 4) Sparse decode (32 live units) ====================
__global__ __launch_bounds__(256) void decode_t_kernel(
    const float* __restrict__ WT, const float* __restrict__ bd,
    const int* __restrict__ tidx, const float* __restrict__ tval,
    float* __restrict__ xhat) {
  __shared__ int   sIdx[K_TOP];
  __shared__ float sVal[K_TOP];
  const int b = blockIdx.x, tid = threadIdx.x;
  if (tid < K_TOP) {
    sIdx[tid] = tidx[(size_t)b * K_TOP + tid];
    sVal[tid] = tval[(size_t)b * K_TOP + tid];
  }
  __syncthreads();
  const int d = blockIdx.y * 256 + tid;
  float acc = bd[d];
#pragma unroll 8
  for (int j = 0; j < K_TOP; ++j)
    acc = fmaf(sVal[j], WT[(size_t)sIdx[j] * D_DIM + d], acc);
  xhat[(size_t)b * D_DIM + d] = acc;
}

__global__ __launch_bounds__(256) void decode_g_kernel(
    const float* __restrict__ Wd, const float* __restrict__ bd,
    const int* __restrict__ tidx, const float* __restrict__ tval,
    float* __restrict__ xhat) {
  __shared__ int   sIdx[K_TOP];
  __shared__ float sVal[K_TOP];
  const int b = blockIdx.x, tid = threadIdx.x;
  if (tid < K_TOP) {
    sIdx[tid] = tidx[(size_t)b * K_TOP + tid];
    sVal[tid] = tval[(size_t)b * K_TOP + tid];
  }
  __syncthreads();
  const int d = blockIdx.y * 256 + tid;
  float acc = bd[d];
#pragma unroll 8
  for (int j = 0; j < K_TOP; ++j)
    acc = fmaf(sVal[j], Wd[(size_t)d * H_DIM + sIdx[j]], acc);
  xhat[(size_t)b * D_DIM + d] = acc;
}

// ================================ launch ===================================
extern "C" void kernel_launch(void* const* d_in, const int* in_sizes, int n_in,
                              void* d_out, int out_size, void* d_ws, size_t ws_size,
                              hipStream_t stream) {
  const float* x  = (const float*)d_in[0];   // [B, D]
  const float* We = (const float*)d_in[1];   // [H, D]
  const float* be = (const float*)d_in[2];   // [H]
  const float* Wd = (const float*)d_in[3];   // [D, H]
  const float* bd = (const float*)d_in[4];   // [D]
  // d_in[5] = k (constant 32, baked in)

  float* xhat = (float*)d_out;
  float* z    = (float*)d_out + (size_t)B_DIM * D_DIM;

  // workspace layout
  const size_t listElems = (size_t)B_DIM * K_TOP;
  int*   tidx = (int*)d_ws;
  float* tval = (float*)((char*)d_ws + listElems * sizeof(int));
  const size_t wdtOff  = listElems * (sizeof(int) + sizeof(float));
  float* WdT = (float*)((char*)d_ws + wdtOff);
  const bool useT = ws_size >= wdtOff + (size_t)H_DIM * D_DIM * sizeof(float);

  // 1) encode GEMM -> unmasked z
  encode_wmma_kernel<<<dim3(B_DIM / TM, H_DIM / TN), 128, 0, stream>>>(x, We, be, z);

  // 3) transpose W_dec (independent; overlaps stream-serially with topk)
  if (useT)
    transpose_kernel<<<dim3(H_DIM / 32, D_DIM / 32), dim3(32, 8), 0, stream>>>(Wd, WdT);

  // 2) top-k mask in place + compact (idx,val) lists
  topk_mask_kernel<<<B_DIM, 256, 0, stream>>>(z, tidx, tval);

  // 4) sparse decode
  if (useT)
    decode_t_kernel<<<dim3(B_DIM, D_DIM / 256), 256, 0, stream>>>(WdT, bd, tidx, tval, xhat);
  else
    decode_g_kernel<<<dim3(B_DIM, D_DIM / 256), 256, 0, stream>>>(Wd, bd, tidx, tval, xhat);
}